// DGFM_43293270343951
// MI455X (gfx1250) — compile-verified
//
#include <hip/hip_runtime.h>
#include <math.h>

// ---------------- CDNA5 WMMA (gfx1250, wave32) ----------------
typedef float v2f __attribute__((ext_vector_type(2)));
typedef float v8f __attribute__((ext_vector_type(8)));

__device__ __forceinline__ v8f wmma4(v2f a, v2f b, v8f c) {
  // D(16x16,f32) = A(16x4,f32) * B(4x16,f32) + C ; full fp32 matrix pipe
  return __builtin_amdgcn_wmma_f32_16x16x4_f32(false, a, false, b, (short)0, c,
                                               false, false);
}

#define HH 192
#define WWd 192
#define HWp (HH * WWd)      // 36864
#define CCh 64
#define BBn 2
#define PTILES (HWp / 16)   // 2304 pixel tiles per batch image

// ---------------- elementwise kernels ----------------
__global__ void add3_kernel(const float* __restrict__ a, const float* __restrict__ b,
                            const float* __restrict__ c, float* __restrict__ o, int n4) {
  int i = blockIdx.x * blockDim.x + threadIdx.x;
  if (i < n4) {
    float4 va = ((const float4*)a)[i];
    float4 vb = ((const float4*)b)[i];
    float4 vc = ((const float4*)c)[i];
    float4 vo;
    vo.x = va.x + vb.x + vc.x;
    vo.y = va.y + vb.y + vc.y;
    vo.z = va.z + vb.z + vc.z;
    vo.w = va.w + vb.w + vc.w;
    ((float4*)o)[i] = vo;
  }
}

__global__ void tanhmul_kernel(const float* __restrict__ g, const float* __restrict__ x,
                               float* __restrict__ o, int n4) {
  int i = blockIdx.x * blockDim.x + threadIdx.x;
  if (i < n4) {
    float4 vg = ((const float4*)g)[i];
    float4 vx = ((const float4*)x)[i];
    float4 vo;
    vo.x = tanhf(vg.x) * vx.x;
    vo.y = tanhf(vg.y) * vx.y;
    vo.z = tanhf(vg.z) * vx.z;
    vo.w = tanhf(vg.w) * vx.w;
    ((float4*)o)[i] = vo;
  }
}

// ---------------- weight repack kernels (tap-major, b64-friendly) ----------------
// wcomb[t][oc][c] : rows [0,2kk) = w_off, [2kk,3kk) = w_mod, rest zero; bcomb likewise.
__global__ void repack_wcomb(const float* __restrict__ w_off, const float* __restrict__ w_mod,
                             const float* __restrict__ b_off, const float* __restrict__ b_mod,
                             float* __restrict__ wcomb, float* __restrict__ bcomb,
                             int kk, int CTR, int total) {
  int idx = blockIdx.x * blockDim.x + threadIdx.x;
  if (idx < total) {
    int c = idx & 63;
    int oc = (idx >> 6) % CTR;
    int t = idx / (64 * CTR);
    float v = 0.0f;
    if (oc < 2 * kk)      v = w_off[((size_t)oc * CCh + c) * kk + t];
    else if (oc < 3 * kk) v = w_mod[((size_t)(oc - 2 * kk) * CCh + c) * kk + t];
    wcomb[idx] = v;
  }
  if (idx < CTR) {
    float bv = 0.0f;
    if (idx < 2 * kk)      bv = b_off[idx];
    else if (idx < 3 * kk) bv = b_mod[idx - 2 * kk];
    bcomb[idx] = bv;
  }
}

// wregT[t][o][c] = w_reg[o][c][t]
__global__ void repack_wreg(const float* __restrict__ w_reg, float* __restrict__ wregT,
                            int kk, int total) {
  int idx = blockIdx.x * blockDim.x + threadIdx.x;
  if (idx < total) {
    int c = idx & 63;
    int o = (idx >> 6) & 63;
    int t = idx >> 12;
    wregT[idx] = w_reg[((size_t)o * CCh + c) * kk + t];
  }
}

// ---------------- 1x1 conv (64x64 GEMM over pixels), WMMA ----------------
// One wave -> 64 out-channels x 16 pixels. TR_IN/TR_OUT select pixel-major [p][c] layout.
template <int RELU, int HAS_RES, int TR_IN, int TR_OUT>
__global__ void conv1x1_wmma(const float* __restrict__ in, const float* __restrict__ w,
                             const float* __restrict__ res, float* __restrict__ out) {
  int lane = threadIdx.x & 31;
  int l16 = lane & 15, half = lane >> 4;
  int wid = blockIdx.x * (blockDim.x >> 5) + (threadIdx.x >> 5);
  int ptile = wid % PTILES;
  int b = wid / PTILES;
  int p = ptile * 16 + l16;
  const float* inb = in + (size_t)b * CCh * HWp;

  v8f acc[4] = {};
#pragma unroll 4
  for (int k0 = 0; k0 < CCh; k0 += 4) {
    int c0 = k0 + 2 * half;
    v2f bf;
    if (TR_IN) {
      bf = *(const v2f*)(inb + (size_t)p * CCh + c0);           // one b64
    } else {
      bf.x = inb[(size_t)c0 * HWp + p];
      bf.y = inb[(size_t)(c0 + 1) * HWp + p];
    }
#pragma unroll
    for (int ot = 0; ot < 4; ot++) {
      int row = ot * 16 + l16;
      v2f af = *(const v2f*)(w + row * CCh + c0);               // one b64
      acc[ot] = wmma4(af, bf, acc[ot]);
    }
  }
  size_t ob = (size_t)b * CCh * HWp;
#pragma unroll
  for (int ot = 0; ot < 4; ot++) {
    if (TR_OUT) {
      float4 lo, hi;
      lo.x = acc[ot][0]; lo.y = acc[ot][1]; lo.z = acc[ot][2]; lo.w = acc[ot][3];
      hi.x = acc[ot][4]; hi.y = acc[ot][5]; hi.z = acc[ot][6]; hi.w = acc[ot][7];
      if (RELU) {
        lo.x = fmaxf(lo.x, 0.f); lo.y = fmaxf(lo.y, 0.f);
        lo.z = fmaxf(lo.z, 0.f); lo.w = fmaxf(lo.w, 0.f);
        hi.x = fmaxf(hi.x, 0.f); hi.y = fmaxf(hi.y, 0.f);
        hi.z = fmaxf(hi.z, 0.f); hi.w = fmaxf(hi.w, 0.f);
      }
      float* dst = out + ob + (size_t)p * CCh + ot * 16 + 8 * half;
      *(float4*)dst = lo;
      *(float4*)(dst + 4) = hi;
    } else {
#pragma unroll
      for (int r = 0; r < 8; r++) {
        int o = ot * 16 + r + 8 * half;
        float v = acc[ot][r];
        if (RELU) v = fmaxf(v, 0.0f);
        if (HAS_RES) v += res[ob + (size_t)o * HWp + p];
        out[ob + (size_t)o * HWp + p] = v;
      }
    }
  }
}

// ---------------- kxk offset+mask conv as implicit per-tap shifted GEMM ----------------
// hT is pixel-major [p][c]; weights pre-packed tap-major in wcomb. One wave -> 16 oc x 16 px.
// K templated: no runtime idiv, constant CT, unrollable channel loop.
template <int K>
__global__ void offmask_conv_wmma(const float* __restrict__ hT,
                                  const float* __restrict__ wcomb, const float* __restrict__ bcomb,
                                  float* __restrict__ off, float* __restrict__ maskb) {
  constexpr int kk = K * K, pad = K / 2;
  constexpr int CT = (3 * kk + 15) / 16, CTR = CT * 16;
  int lane = threadIdx.x & 31;
  int l16 = lane & 15, half = lane >> 4;
  int wid = blockIdx.x * (blockDim.x >> 5) + (threadIdx.x >> 5);
  int octile = wid % CT;
  int rest = wid / CT;
  int ptile = rest % PTILES;
  int b = rest / PTILES;
  int p0 = ptile * 16;
  int y = p0 / WWd;              // uniform across tile (192 % 16 == 0)
  int x = (p0 % WWd) + l16;
  int oc = octile * 16 + l16;
  const float* hTb = hT + (size_t)b * CCh * HWp;

  v8f acc = {};
#pragma unroll 1
  for (int t = 0; t < kk; t++) {
    int di = t / K - pad, dj = t % K - pad;
    int yy = y + di, xx = x + dj;
    bool inb = (yy >= 0) && (yy < HH) && (xx >= 0) && (xx < WWd);
    const float* src = hTb + (size_t)(inb ? (yy * WWd + xx) : 0) * CCh;
    const float* wsrc = wcomb + ((size_t)t * CTR + oc) * CCh;
#pragma unroll 4
    for (int k0 = 0; k0 < CCh; k0 += 4) {
      int c0 = k0 + 2 * half;
      v2f bf = {0.0f, 0.0f};
      if (inb) bf = *(const v2f*)(src + c0);                    // one b64 (exec-masked)
      v2f af = *(const v2f*)(wsrc + c0);                        // one b64, no branch
      acc = wmma4(af, bf, acc);
    }
  }
  int p = p0 + l16;
#pragma unroll
  for (int r = 0; r < 8; r++) {
    int o = octile * 16 + r + 8 * half;
    float v = acc[r] + bcomb[o];
    if (o < 2 * kk) {
      off[((size_t)b * 2 * kk + o) * HWp + p] = v;
    } else if (o < 3 * kk) {
      maskb[((size_t)b * kk + (o - 2 * kk)) * HWp + p] = 2.0f / (1.0f + expf(-v));
    }
  }
}

// ---------------- modulated deformable conv: bilinear gather + per-tap GEMM ----------------
// hT pixel-major; wregT tap-major. One wave -> 64 oc x 16 px; output pixel-major.
template <int K>
__global__ void deform_conv_wmma(const float* __restrict__ hT,
                                 const float* __restrict__ off, const float* __restrict__ maskb,
                                 const float* __restrict__ wregT, float* __restrict__ out) {
  constexpr int kk = K * K, pad = K / 2;
  int lane = threadIdx.x & 31;
  int l16 = lane & 15, half = lane >> 4;
  int wid = blockIdx.x * (blockDim.x >> 5) + (threadIdx.x >> 5);
  int ptile = wid % PTILES;
  int b = wid / PTILES;
  int p = ptile * 16 + l16;
  int y = p / WWd, x = p % WWd;
  const float* hTb = hT + (size_t)b * CCh * HWp;

  v8f acc[4] = {};
#pragma unroll 1
  for (int t = 0; t < kk; t++) {
    float oy = off[((size_t)b * 2 * kk + 2 * t) * HWp + p];
    float ox = off[((size_t)b * 2 * kk + 2 * t + 1) * HWp + p];
    float mv = maskb[((size_t)b * kk + t) * HWp + p];
    float py = (float)(y - pad + t / K) + oy;
    float px = (float)(x - pad + t % K) + ox;
    float fy = floorf(py), fx = floorf(px);
    int iy0 = (int)fy, ix0 = (int)fx;
    int iy1 = iy0 + 1, ix1 = ix0 + 1;
    float wy1 = py - fy, wx1 = px - fx;
    float wy0 = 1.0f - wy1, wx0 = 1.0f - wx1;
    float v00 = (iy0 >= 0 && iy0 < HH && ix0 >= 0 && ix0 < WWd) ? 1.0f : 0.0f;
    float v01 = (iy0 >= 0 && iy0 < HH && ix1 >= 0 && ix1 < WWd) ? 1.0f : 0.0f;
    float v10 = (iy1 >= 0 && iy1 < HH && ix0 >= 0 && ix0 < WWd) ? 1.0f : 0.0f;
    float v11 = (iy1 >= 0 && iy1 < HH && ix1 >= 0 && ix1 < WWd) ? 1.0f : 0.0f;
    int cy0 = min(max(iy0, 0), HH - 1), cy1 = min(max(iy1, 0), HH - 1);
    int cx0 = min(max(ix0, 0), WWd - 1), cx1 = min(max(ix1, 0), WWd - 1);
    const float* r00 = hTb + (size_t)(cy0 * WWd + cx0) * CCh;
    const float* r01 = hTb + (size_t)(cy0 * WWd + cx1) * CCh;
    const float* r10 = hTb + (size_t)(cy1 * WWd + cx0) * CCh;
    const float* r11 = hTb + (size_t)(cy1 * WWd + cx1) * CCh;
    float W00 = wy0 * wx0 * mv * v00, W01 = wy0 * wx1 * mv * v01;
    float W10 = wy1 * wx0 * mv * v10, W11 = wy1 * wx1 * mv * v11;
    const float* wt = wregT + (size_t)t * (CCh * CCh);

#pragma unroll 4
    for (int k0 = 0; k0 < CCh; k0 += 4) {
      int c0 = k0 + 2 * half;
      v2f s00 = *(const v2f*)(r00 + c0);                        // 4 x b64 gathers
      v2f s01 = *(const v2f*)(r01 + c0);
      v2f s10 = *(const v2f*)(r10 + c0);
      v2f s11 = *(const v2f*)(r11 + c0);
      v2f bf = W00 * s00 + W01 * s01 + W10 * s10 + W11 * s11;
#pragma unroll
      for (int ot = 0; ot < 4; ot++) {
        int row = ot * 16 + l16;
        v2f af = *(const v2f*)(wt + row * CCh + c0);            // one b64
        acc[ot] = wmma4(af, bf, acc[ot]);
      }
    }
  }
  // pixel-major output: [p][o], 8 consecutive channels per lane -> b128 stores
  float* ob = out + (size_t)b * CCh * HWp + (size_t)p * CCh;
#pragma unroll
  for (int ot = 0; ot < 4; ot++) {
    float4 lo, hi;
    lo.x = acc[ot][0]; lo.y = acc[ot][1]; lo.z = acc[ot][2]; lo.w = acc[ot][3];
    hi.x = acc[ot][4]; hi.y = acc[ot][5]; hi.z = acc[ot][6]; hi.w = acc[ot][7];
    float* dst = ob + ot * 16 + 8 * half;
    *(float4*)dst = lo;
    *(float4*)(dst + 4) = hi;
  }
}

// ---------------- launch ----------------
extern "C" void kernel_launch(void* const* d_in, const int* in_sizes, int n_in,
                              void* d_out, int out_size, void* d_ws, size_t ws_size,
                              hipStream_t stream) {
  (void)in_sizes; (void)n_in; (void)out_size; (void)ws_size;
  const float* x_in[3] = {(const float*)d_in[0], (const float*)d_in[1], (const float*)d_in[2]};
  const float* w1 = (const float*)d_in[21];
  const float* w2 = (const float*)d_in[22];

  float* ws = (float*)d_ws;
  const size_t S = (size_t)BBn * CCh * HWp;          // 4,718,592 floats
  float* y   = ws;                                   // residual (NCHW)
  float* a   = ws + 1 * S;                           // branch 'max' out / t / t2 (pixel-major)
  float* m   = ws + 2 * S;                           // branch 'mid' out / u (pixel-major)
  float* s3  = ws + 3 * S;                           // branch 'small' out (pixel-major)
  float* hT  = ws + 4 * S;                           // pre-conv activations, pixel-major [p][c]
  float* off = ws + 5 * S;                           // cap k=7: 98*B*HW (channel-major)
  float* mk  = off + (size_t)98 * BBn * HWp;         // cap k=7: 49*B*HW
  float* wcomb = mk + (size_t)49 * BBn * HWp;        // cap 49*160*64
  float* bcomb = wcomb + (size_t)49 * 160 * CCh;     // cap 160
  float* wregT = bcomb + 160;                        // cap 49*64*64

  dim3 blk(256);                                     // 8 wave32 per block
  const int n4 = (int)(S / 4);
  const int ewGrid = (n4 + 255) / 256;
  const int gemmGrid = BBn * PTILES / 8;             // 576 blocks, exact

  add3_kernel<<<ewGrid, blk, 0, stream>>>(x_in[0], x_in[1], x_in[2], y, n4);

  const int ks[3] = {7, 5, 3};
  float* bout[3] = {a, m, s3};
  for (int br = 0; br < 3; br++) {
    int base = 3 + br * 6;
    const float* w_pre = (const float*)d_in[base + 0];
    const float* w_off = (const float*)d_in[base + 1];
    const float* b_off = (const float*)d_in[base + 2];
    const float* w_mod = (const float*)d_in[base + 3];
    const float* b_mod = (const float*)d_in[base + 4];
    const float* w_reg = (const float*)d_in[base + 5];
    int k = ks[br], kk = k * k;
    int CT = (3 * kk + 15) / 16, CTR = CT * 16;

    int totC = kk * CTR * CCh;
    repack_wcomb<<<(totC + 255) / 256, blk, 0, stream>>>(w_off, w_mod, b_off, b_mod,
                                                         wcomb, bcomb, kk, CTR, totC);
    int totR = kk * CCh * CCh;
    repack_wreg<<<(totR + 255) / 256, blk, 0, stream>>>(w_reg, wregT, kk, totR);

    // pre: NCHW in -> pixel-major hT, relu
    conv1x1_wmma<1, 0, 0, 1><<<gemmGrid, blk, 0, stream>>>(x_in[br], w_pre, nullptr, hT);

    int omGrid = BBn * PTILES * CT / 8;
    switch (k) {
      case 7:
        offmask_conv_wmma<7><<<omGrid, blk, 0, stream>>>(hT, wcomb, bcomb, off, mk);
        deform_conv_wmma<7><<<gemmGrid, blk, 0, stream>>>(hT, off, mk, wregT, bout[br]);
        break;
      case 5:
        offmask_conv_wmma<5><<<omGrid, blk, 0, stream>>>(hT, wcomb, bcomb, off, mk);
        deform_conv_wmma<5><<<gemmGrid, blk, 0, stream>>>(hT, off, mk, wregT, bout[br]);
        break;
      default:
        offmask_conv_wmma<3><<<omGrid, blk, 0, stream>>>(hT, wcomb, bcomb, off, mk);
        deform_conv_wmma<3><<<gemmGrid, blk, 0, stream>>>(hT, off, mk, wregT, bout[br]);
        break;
    }
  }

  // x = tanh(m)*a ; x = relu(w1@x) ; x = tanh(s)*x ; out = relu(w2@x) + y
  tanhmul_kernel<<<ewGrid, blk, 0, stream>>>(m, a, a, n4);
  conv1x1_wmma<1, 0, 1, 1><<<gemmGrid, blk, 0, stream>>>(a, w1, nullptr, m);   // pixel-major u
  tanhmul_kernel<<<ewGrid, blk, 0, stream>>>(s3, m, a, n4);
  conv1x1_wmma<1, 1, 1, 0><<<gemmGrid, blk, 0, stream>>>(a, w2, y, (float*)d_out); // NCHW + res
}